// EqStructGNN_78176994722114
// MI455X (gfx1250) — compile-verified
//
#include <hip/hip_runtime.h>

#define NUU 100000
#define NII 50000
#define EE  500000
#define CC  128

typedef __attribute__((ext_vector_type(2))) float v2f;
typedef __attribute__((ext_vector_type(8))) float v8f;

__device__ __forceinline__ float wave_sum(float v) {
#pragma unroll
  for (int m = 16; m >= 1; m >>= 1) v += __shfl_xor(v, m, 32);
  return v;
}

// Y[rows x 128] = A1 @ W1 (+ A2 @ W2) + bias, optional relu.
// One 16x16 output tile per wave via V_WMMA_F32_16X16X4_F32; 8 waves/block
// cover all 8 column tiles of C=128 for one 16-row slab. rows % 16 == 0.
__global__ __launch_bounds__(256) void gemm_dual(
    const float* __restrict__ A1, const float* __restrict__ W1,
    const float* __restrict__ A2, const float* __restrict__ W2,
    const float* __restrict__ bias, float* __restrict__ Y, int relu)
{
  const int lane = threadIdx.x & 31;
  const int wave = threadIdx.x >> 5;
  const int half = lane >> 4;       // lanes 16-31 hold K+2 slices
  const int l15  = lane & 15;
  const int rowBase = blockIdx.x * 16;
  const int col = wave * 16 + l15;  // output column
  const int m   = rowBase + l15;    // A row for this lane

  v8f acc = {};
  const float* a1 = A1 + (size_t)m * CC;
#pragma unroll
  for (int k0 = 0; k0 < CC; k0 += 4) {
    const int kk = k0 + 2 * half;
    v2f a; a.x = a1[kk];               a.y = a1[kk + 1];
    v2f b; b.x = W1[kk * CC + col];    b.y = W1[(kk + 1) * CC + col];
    acc = __builtin_amdgcn_wmma_f32_16x16x4_f32(false, a, false, b,
                                                (short)0, acc, false, false);
  }
  if (A2) {
    const float* a2 = A2 + (size_t)m * CC;
#pragma unroll
    for (int k0 = 0; k0 < CC; k0 += 4) {
      const int kk = k0 + 2 * half;
      v2f a; a.x = a2[kk];             a.y = a2[kk + 1];
      v2f b; b.x = W2[kk * CC + col];  b.y = W2[(kk + 1) * CC + col];
      acc = __builtin_amdgcn_wmma_f32_16x16x4_f32(false, a, false, b,
                                                  (short)0, acc, false, false);
    }
  }
  const float bn = bias ? bias[col] : 0.0f;
#pragma unroll
  for (int v = 0; v < 8; ++v) {
    float val = acc[v] + bn;
    if (relu) val = fmaxf(val, 0.0f);
    Y[(size_t)(rowBase + v + 8 * half) * CC + col] = val;
  }
}

// inv-norm per row: 1 / max(||x||, 1e-8); one wave per row
__global__ __launch_bounds__(256) void rownorm_inv(
    const float* __restrict__ X, float* __restrict__ out, int nrows)
{
  const int r = (int)((blockIdx.x * blockDim.x + threadIdx.x) >> 5);
  const int lane = threadIdx.x & 31;
  if (r >= nrows) return;
  float4 v = ((const float4*)(X + (size_t)r * CC))[lane];
  float s = wave_sum(v.x * v.x + v.y * v.y + v.z * v.z + v.w * v.w);
  if (lane == 0) out[r] = 1.0f / fmaxf(sqrtf(s), 1e-8f);
}

// structural contrastive loss; one wave per edge; hu/hi resident in L2
__global__ __launch_bounds__(256) void struct_loss_k(
    const float* __restrict__ hu, const float* __restrict__ hi,
    const float* __restrict__ nu_inv, const float* __restrict__ ni_inv,
    const int* __restrict__ src, const int* __restrict__ dst,
    const int* __restrict__ unidx, const int* __restrict__ inidx,
    float* __restrict__ loss)
{
  const int e = (int)((blockIdx.x * blockDim.x + threadIdx.x) >> 5);
  const int lane = threadIdx.x & 31;
  if (e >= EE) return;
  const int s = src[e], d = dst[e], un = unidx[e], in = inidx[e];
  float4 ue = ((const float4*)(hu + (size_t)s  * CC))[lane];
  float4 ie = ((const float4*)(hi + (size_t)d  * CC))[lane];
  float4 vn = ((const float4*)(hu + (size_t)un * CC))[lane];
  float4 jn = ((const float4*)(hi + (size_t)in * CC))[lane];
  float dp = ue.x * ie.x + ue.y * ie.y + ue.z * ie.z + ue.w * ie.w;
  float d1 = ue.x * jn.x + ue.y * jn.y + ue.z * jn.z + ue.w * jn.w;
  float d2 = vn.x * ie.x + vn.y * ie.y + vn.z * ie.z + vn.w * ie.w;
  dp = wave_sum(dp); d1 = wave_sum(d1); d2 = wave_sum(d2);
  if (lane == 0) {
    const float pos = dp * nu_inv[s] * ni_inv[d];
    const float neg = d1 * nu_inv[s] * ni_inv[in] + d2 * nu_inv[un] * ni_inv[d];
    const float pe = expf(pos * 10.0f);  // 1/TEMP = 10
    const float ne = expf(neg * 10.0f);
    atomicAdd(loss, -logf(pe / (pe + ne)) * (1.0f / EE));
  }
}

// action head + gumbel hard (forward value is pure argmax one-hot, col 0)
__global__ __launch_bounds__(256) void user_head(
    const float* __restrict__ eu, const float* __restrict__ Wact,
    const float* __restrict__ bact, const float* __restrict__ uu,
    float* __restrict__ probs)
{
  const int u = (int)((blockIdx.x * blockDim.x + threadIdx.x) >> 5);
  const int lane = threadIdx.x & 31;
  if (u >= NUU) return;
  float4 e = ((const float4*)(eu + (size_t)u * CC))[lane];
  const int k = lane * 4;
  float d0 = e.x * Wact[(k + 0) * 2] + e.y * Wact[(k + 1) * 2]
           + e.z * Wact[(k + 2) * 2] + e.w * Wact[(k + 3) * 2];
  float d1 = e.x * Wact[(k + 0) * 2 + 1] + e.y * Wact[(k + 1) * 2 + 1]
           + e.z * Wact[(k + 2) * 2 + 1] + e.w * Wact[(k + 3) * 2 + 1];
  d0 = wave_sum(d0); d1 = wave_sum(d1);
  if (lane == 0) {
    const float g0 = -logf(-logf(uu[u * 2 + 0] * (1.0f - 2e-6f) + 1e-6f));
    const float g1 = -logf(-logf(uu[u * 2 + 1] * (1.0f - 2e-6f) + 1e-6f));
    probs[u] = ((d0 + bact[0] + g0) >= (d1 + bact[1] + g1)) ? 1.0f : 0.0f;
  }
}

// segment_sum scatter: one wave per edge, 4 floats per lane, f32 atomics (L2)
__global__ __launch_bounds__(256) void scatter_add(
    const float* __restrict__ feat, const int* __restrict__ sidx,
    const int* __restrict__ didx, float* __restrict__ agg, int nE)
{
  const int e = (int)((blockIdx.x * blockDim.x + threadIdx.x) >> 5);
  const int lane = threadIdx.x & 31;
  if (e >= nE) return;
  const int s = sidx[e], d = didx[e];
  float4 v = ((const float4*)(feat + (size_t)s * CC))[lane];
  float* dst = agg + (size_t)d * CC + lane * 4;
  atomicAdd(dst + 0, v.x); atomicAdd(dst + 1, v.y);
  atomicAdd(dst + 2, v.z); atomicAdd(dst + 3, v.w);
}

// in-place LayerNorm over C=128; one wave per row
__global__ __launch_bounds__(256) void layernorm(
    float* __restrict__ X, const float* __restrict__ g,
    const float* __restrict__ b, int nrows)
{
  const int r = (int)((blockIdx.x * blockDim.x + threadIdx.x) >> 5);
  const int lane = threadIdx.x & 31;
  if (r >= nrows) return;
  float4* row = (float4*)(X + (size_t)r * CC);
  float4 v = row[lane];
  float s = wave_sum(v.x + v.y + v.z + v.w);
  float q = wave_sum(v.x * v.x + v.y * v.y + v.z * v.z + v.w * v.w);
  const float mean = s * (1.0f / CC);
  const float var  = q * (1.0f / CC) - mean * mean;
  const float rstd = rsqrtf(var + 1e-5f);
  const int c = lane * 4;
  v.x = (v.x - mean) * rstd * g[c + 0] + b[c + 0];
  v.y = (v.y - mean) * rstd * g[c + 1] + b[c + 1];
  v.z = (v.z - mean) * rstd * g[c + 2] + b[c + 2];
  v.w = (v.w - mean) * rstd * g[c + 3] + b[c + 3];
  row[lane] = v;
}

extern "C" void kernel_launch(void* const* d_in, const int* in_sizes, int n_in,
                              void* d_out, int out_size, void* d_ws, size_t ws_size,
                              hipStream_t stream) {
  (void)in_sizes; (void)n_in; (void)out_size; (void)ws_size;
  const float* x_user  = (const float*)d_in[0];
  const float* x_item  = (const float*)d_in[1];
  const float* u_user  = (const float*)d_in[3];
  const float* Wu_proj = (const float*)d_in[5];
  const float* bu_proj = (const float*)d_in[6];
  const float* Wi_proj = (const float*)d_in[7];
  const float* bi_proj = (const float*)d_in[8];
  const float* Wu_enc  = (const float*)d_in[9];
  const float* bu_enc  = (const float*)d_in[10];
  const float* Wu_act  = (const float*)d_in[11];
  const float* bu_act  = (const float*)d_in[12];
  const float* Wl_ui   = (const float*)d_in[17];
  const float* Wr_ui   = (const float*)d_in[18];
  const float* b_ui    = (const float*)d_in[19];
  const float* Wl_iu   = (const float*)d_in[20];
  const float* Wr_iu   = (const float*)d_in[21];
  const float* b_iu    = (const float*)d_in[22];
  const float* ln_u_g  = (const float*)d_in[23];
  const float* ln_u_b  = (const float*)d_in[24];
  const float* ln_i_g  = (const float*)d_in[25];
  const float* ln_i_b  = (const float*)d_in[26];
  const int*   edge_ui = (const int*)d_in[27];   // [2,E] row0=users row1=items
  const int*   edge_iu = (const int*)d_in[28];   // [2,E] row0=items row1=users
  const int*   user_neg = (const int*)d_in[30];
  const int*   item_neg = (const int*)d_in[31];

  float* out = (float*)d_out;
  const size_t o_hi   = (size_t)NUU * CC;
  const size_t o_loss = o_hi + (size_t)NII * CC;
  float* U_b   = out;               // final hu lives here
  float* I_b   = out + o_hi;        // final hi lives here
  float* loss  = out + o_loss;
  float* probs = out + o_loss + 1;

  float* ws     = (float*)d_ws;
  float* U_a    = ws;                         // NU*C
  float* aggU   = U_a  + (size_t)NUU * CC;    // NU*C (doubles as eu)
  float* I_a    = aggU + (size_t)NUU * CC;    // NI*C
  float* aggI   = I_a  + (size_t)NII * CC;    // NI*C
  float* nu_inv = aggI + (size_t)NII * CC;    // NU
  float* ni_inv = nu_inv + NUU;               // NI

  const dim3 blk(256);
  // projections (fp32 WMMA GEMMs)
  gemm_dual<<<NUU / 16, blk, 0, stream>>>(x_user, Wu_proj, nullptr, nullptr, bu_proj, U_a, 0);
  gemm_dual<<<NII / 16, blk, 0, stream>>>(x_item, Wi_proj, nullptr, nullptr, bi_proj, I_a, 0);
  // user encoder (relu) -> eu (reuses aggU), then gumbel-hard head
  gemm_dual<<<NUU / 16, blk, 0, stream>>>(U_a, Wu_enc, nullptr, nullptr, bu_enc, aggU, 1);
  user_head<<<NUU * 32 / 256, blk, 0, stream>>>(aggU, Wu_act, bu_act, u_user, probs);
  // cosine norms + contrastive loss over positive edges
  rownorm_inv<<<NUU * 32 / 256, blk, 0, stream>>>(U_a, nu_inv, NUU);
  rownorm_inv<<<NII * 32 / 256, blk, 0, stream>>>(I_a, ni_inv, NII);
  hipMemsetAsync(loss, 0, sizeof(float), stream);
  struct_loss_k<<<EE * 32 / 256, blk, 0, stream>>>(U_a, I_a, nu_inv, ni_inv,
                                                   edge_ui, edge_ui + EE,
                                                   user_neg, item_neg, loss);
  // 3 hetero SAGE layers; ping-pong so layer 2 writes directly into d_out
  const float* inU = U_a;
  const float* inI = I_a;
  for (int l = 0; l < 3; ++l) {
    float* outU = (l & 1) ? U_a : U_b;
    float* outI = (l & 1) ? I_a : I_b;
    hipMemsetAsync(aggI, 0, (size_t)NII * CC * sizeof(float), stream);
    hipMemsetAsync(aggU, 0, (size_t)NUU * CC * sizeof(float), stream);
    scatter_add<<<EE * 32 / 256, blk, 0, stream>>>(inU, edge_ui, edge_ui + EE, aggI, EE);
    scatter_add<<<EE * 32 / 256, blk, 0, stream>>>(inI, edge_iu, edge_iu + EE, aggU, EE);
    gemm_dual<<<NII / 16, blk, 0, stream>>>(aggI, Wl_ui + (size_t)l * CC * CC,
                                            inI,  Wr_ui + (size_t)l * CC * CC,
                                            b_ui + l * CC, outI, 0);
    gemm_dual<<<NUU / 16, blk, 0, stream>>>(aggU, Wl_iu + (size_t)l * CC * CC,
                                            inU,  Wr_iu + (size_t)l * CC * CC,
                                            b_iu + l * CC, outU, 0);
    if (l < 2) {
      layernorm<<<NUU * 32 / 256, blk, 0, stream>>>(outU, ln_u_g + l * CC, ln_u_b + l * CC, NUU);
      layernorm<<<NII * 32 / 256, blk, 0, stream>>>(outI, ln_i_g + l * CC, ln_i_b + l * CC, NII);
    }
    inU = outU; inI = outI;
  }
}